// GNNLayer_2619930051568
// MI455X (gfx1250) — compile-verified
//
#include <hip/hip_runtime.h>

typedef __attribute__((ext_vector_type(2))) float v2f;
typedef __attribute__((ext_vector_type(8))) float v8f;

#define D 128
#define DP (D + 4)        // padded LDS row (bank-conflict-free, keeps 16B align)
#define EPS 1e-5f

__device__ __forceinline__ v8f wmma_f32_k4(v2f a, v2f b, v8f c) {
  // D = A(16x4,f32) * B(4x16,f32) + C(16x16,f32)
  return __builtin_amdgcn_wmma_f32_16x16x4_f32(
      /*neg_a=*/false, a, /*neg_b=*/false, b,
      /*c_mod=*/(short)0, c, /*reuse_a=*/false, /*reuse_b=*/false);
}

// ---------------------------------------------------------------- zero
__global__ void gnn_zero_kernel(float* __restrict__ p, long n) {
  long i = (long)blockIdx.x * blockDim.x + threadIdx.x;
  if (i < n) p[i] = 0.f;
}

// ---------------------------------------------------------------- node GEMM
// block = 128 threads (4 waves), 16 rows of h. Wave w computes matrix w of
// {U,V,B,C}: out[w][row][n] = sum_k h[row][k] * W_w[n][k] + b_w[n]
__global__ __launch_bounds__(128) void gnn_node_gemm_kernel(
    const float* __restrict__ h_in,
    const float* __restrict__ Uw, const float* __restrict__ Vw,
    const float* __restrict__ Bw, const float* __restrict__ Cw,
    const float* __restrict__ Ub, const float* __restrict__ Vb,
    const float* __restrict__ Bb, const float* __restrict__ Cb,
    float* __restrict__ out,  // [4][N][D] contiguous (Uh,Vh,Bh,Ch)
    int nrows) {
  __shared__ float htile[16][DP];
  const int tid = threadIdx.x;
  const int wave = tid >> 5, lane = tid & 31;
  const long row0 = (long)blockIdx.x * 16;

  // stage 16x128 h tile (float4, coalesced)
  for (int i = tid; i < 16 * (D / 4); i += 128) {
    int rr = i / (D / 4), cc = i % (D / 4);
    *(float4*)&htile[rr][cc * 4] = *(const float4*)(h_in + (row0 + rr) * D + cc * 4);
  }
  __syncthreads();

  const float* Wm = (wave == 0) ? Uw : (wave == 1) ? Vw : (wave == 2) ? Bw : Cw;
  const float* bm = (wave == 0) ? Ub : (wave == 1) ? Vb : (wave == 2) ? Bb : Cb;

  const int m = lane & 15;
  const int koff = (lane >> 4) << 1;   // 0 or 2
  const int mhalf = (lane >> 4) << 3;  // 0 or 8

  for (int t = 0; t < 8; ++t) {        // 8 N-tiles of 16 cols
    const int n = t * 16 + m;
    const float* wrow = Wm + (long)n * D;  // W row n (L2-hot)
    v8f c = {};
    for (int k = 0; k < D; k += 4) {
      v2f a = *(const v2f*)&htile[m][k + koff];
      v2f b = *(const v2f*)(wrow + k + koff);
      c = wmma_f32_k4(a, b, c);
    }
    const float bias = bm[n];
    float* obase = out + (long)wave * nrows * D;
    for (int r = 0; r < 8; ++r) {
      obase[(row0 + r + mhalf) * D + n] = c[r] + bias;
    }
  }
}

// ---------------------------------------------------------------- edge kernel
// block = 256 threads (8 waves), 32 edges. Computes Ae via WMMA, fuses bias +
// Bh[dst] + Ch[src] -> pre-BN e (into d_out), accumulates e column stats, and
// does the gated message scatter-add into agg.
__global__ __launch_bounds__(256) void gnn_edge_kernel(
    const float* __restrict__ e_in,
    const long long* __restrict__ eidx,  // [2][E] int64 (row0=dst, row1=src)
    const float* __restrict__ Aw, const float* __restrict__ Ab,
    const float* __restrict__ Vh, const float* __restrict__ Bh,
    const float* __restrict__ Ch,
    float* __restrict__ agg,      // [N][D]
    float* __restrict__ e_stats,  // [2][D]: sum, sumsq
    float* __restrict__ e_pre,    // d_out e region [E][D]
    int nedges) {
  __shared__ float etile[32][DP];
  __shared__ int sdst[32], ssrc[32];
  __shared__ float ssum[D], ssq[D];

  const int tid = threadIdx.x;
  const int wave = tid >> 5, lane = tid & 31;
  const long e0 = (long)blockIdx.x * 32;

  for (int i = tid; i < 32 * (D / 4); i += 256) {
    int rr = i / (D / 4), cc = i % (D / 4);
    *(float4*)&etile[rr][cc * 4] = *(const float4*)(e_in + (e0 + rr) * D + cc * 4);
  }
  if (tid < 32) {
    sdst[tid] = (int)eidx[e0 + tid];
    ssrc[tid] = (int)eidx[(long)nedges + e0 + tid];
  }
  if (tid < D) { ssum[tid] = 0.f; ssq[tid] = 0.f; }
  __syncthreads();

  // gated message + segment-sum (atomic scatter into L2-resident agg)
  for (int i = tid; i < 32 * D; i += 256) {
    const int r = i >> 7, c = i & (D - 1);
    const float ev = etile[r][c];
    const float gate = 1.f / (1.f + __expf(-ev));
    const float msg = gate * Vh[(long)ssrc[r] * D + c];
    atomicAdd(&agg[(long)sdst[r] * D + c], msg);
  }

  // Ae = e_tile @ Aw^T via WMMA; waves 0-3 -> edge rows 0-15, waves 4-7 -> 16-31
  const int mtile = wave >> 2;
  const int m = lane & 15;
  const int koff = (lane >> 4) << 1;
  const int mhalf = (lane >> 4) << 3;

  for (int t = 0; t < 2; ++t) {
    const int n = ((wave & 3) * 2 + t) * 16 + m;
    const float* wrow = Aw + (long)n * D;
    v8f c = {};
    for (int k = 0; k < D; k += 4) {
      v2f a = *(const v2f*)&etile[mtile * 16 + m][k + koff];
      v2f b = *(const v2f*)(wrow + k + koff);
      c = wmma_f32_k4(a, b, c);
    }
    const float bias = Ab[n];
    float lsum = 0.f, lsq = 0.f;
    for (int r = 0; r < 8; ++r) {
      const int mm = mtile * 16 + r + mhalf;
      const float val = c[r] + bias + Bh[(long)sdst[mm] * D + n] +
                        Ch[(long)ssrc[mm] * D + n];
      e_pre[(e0 + mm) * D + n] = val;
      lsum += val;
      lsq += val * val;
    }
    atomicAdd(&ssum[n], lsum);  // ds_add_f32
    atomicAdd(&ssq[n], lsq);
  }
  __syncthreads();
  if (tid < D) {
    atomicAdd(&e_stats[tid], ssum[tid]);
    atomicAdd(&e_stats[D + tid], ssq[tid]);
  }
}

// ---------------------------------------------------------------- node post
__global__ __launch_bounds__(256) void gnn_node_post_kernel(
    const float* __restrict__ Uh, const float* __restrict__ agg,
    float* __restrict__ h_pre, float* __restrict__ h_stats, int nrows) {
  __shared__ float ssum[D], ssq[D];
  const int tid = threadIdx.x;
  if (tid < D) { ssum[tid] = 0.f; ssq[tid] = 0.f; }
  __syncthreads();
  const int c = tid & (D - 1);
  const int rsub = tid >> 7;  // 0 or 1
  const long row0 = (long)blockIdx.x * 16;
  float lsum = 0.f, lsq = 0.f;
  for (int r = rsub; r < 16; r += 2) {
    const long off = (row0 + r) * D + c;
    const float v = Uh[off] + agg[off];
    h_pre[off] = v;
    lsum += v;
    lsq += v * v;
  }
  atomicAdd(&ssum[c], lsum);
  atomicAdd(&ssq[c], lsq);
  __syncthreads();
  if (tid < D) {
    atomicAdd(&h_stats[tid], ssum[tid]);
    atomicAdd(&h_stats[D + tid], ssq[tid]);
  }
}

// ---------------------------------------------------------------- finalize BN
__global__ void gnn_stats_final_kernel(
    const float* __restrict__ h_stats, const float* __restrict__ e_stats,
    const float* __restrict__ h_gamma, const float* __restrict__ h_beta,
    const float* __restrict__ e_gamma, const float* __restrict__ e_beta,
    float* __restrict__ h_sc, float* __restrict__ h_sh,
    float* __restrict__ e_sc, float* __restrict__ e_sh, int nrows, int nedges) {
  const int tid = threadIdx.x;
  if (tid < D) {
    const float inv = 1.f / (float)nrows;
    const float mean = h_stats[tid] * inv;
    const float var = h_stats[D + tid] * inv - mean * mean;
    const float sc = h_gamma[tid] * rsqrtf(var + EPS);
    h_sc[tid] = sc;
    h_sh[tid] = h_beta[tid] - mean * sc;
  } else if (tid < 2 * D) {
    const int c = tid - D;
    const float inv = 1.f / (float)nedges;
    const float mean = e_stats[c] * inv;
    const float var = e_stats[D + c] * inv - mean * mean;
    const float sc = e_gamma[c] * rsqrtf(var + EPS);
    e_sc[c] = sc;
    e_sh[c] = e_beta[c] - mean * sc;
  }
}

// ---------------------------------------------------------------- apply
// out = resid + relu(pre * scale + shift), in place over d_out, float4.
__global__ __launch_bounds__(256) void gnn_apply_kernel(
    const float* __restrict__ h_in, const float* __restrict__ e_in,
    const float* __restrict__ h_sc, const float* __restrict__ h_sh,
    const float* __restrict__ e_sc, const float* __restrict__ e_sh,
    float* __restrict__ out, long nh, long ne) {
  const long i4 = (long)blockIdx.x * blockDim.x + threadIdx.x;
  if (i4 >= (nh + ne) / 4) return;
  const long i = i4 * 4;
  const float* resid;
  const float* sc;
  const float* sh;
  long j;
  if (i < nh) { resid = h_in; sc = h_sc; sh = h_sh; j = i; }
  else        { resid = e_in; sc = e_sc; sh = e_sh; j = i - nh; }
  const int c = (int)(j & (D - 1));
  const float4 v = *(const float4*)(out + i);
  const float4 r = *(const float4*)(resid + j);
  float4 o;
  o.x = r.x + fmaxf(fmaf(v.x, sc[c + 0], sh[c + 0]), 0.f);
  o.y = r.y + fmaxf(fmaf(v.y, sc[c + 1], sh[c + 1]), 0.f);
  o.z = r.z + fmaxf(fmaf(v.z, sc[c + 2], sh[c + 2]), 0.f);
  o.w = r.w + fmaxf(fmaf(v.w, sc[c + 3], sh[c + 3]), 0.f);
  *(float4*)(out + i) = o;
}

extern "C" void kernel_launch(void* const* d_in, const int* in_sizes, int n_in,
                              void* d_out, int out_size, void* d_ws, size_t ws_size,
                              hipStream_t stream) {
  const int N = in_sizes[0] / D;
  const int E = in_sizes[1] / D;

  const float* h_in = (const float*)d_in[0];
  const float* e_in = (const float*)d_in[1];
  const long long* eidx = (const long long*)d_in[2];
  const float* Uw = (const float*)d_in[3];
  const float* Ub = (const float*)d_in[4];
  const float* Vw = (const float*)d_in[5];
  const float* Vb = (const float*)d_in[6];
  const float* Aw = (const float*)d_in[7];
  const float* Ab = (const float*)d_in[8];
  const float* Bw = (const float*)d_in[9];
  const float* Bb = (const float*)d_in[10];
  const float* Cw = (const float*)d_in[11];
  const float* Cb = (const float*)d_in[12];
  const float* h_gamma = (const float*)d_in[13];
  const float* h_beta = (const float*)d_in[14];
  const float* e_gamma = (const float*)d_in[15];
  const float* e_beta = (const float*)d_in[16];

  const long ND = (long)N * D;
  const long ED = (long)E * D;

  float* ws = (float*)d_ws;
  float* Uh = ws;             // [N][D]
  float* Vh = Uh + ND;        // [N][D]
  float* Bh = Vh + ND;        // [N][D]
  float* Ch = Bh + ND;        // [N][D]
  float* agg = Ch + ND;       // [N][D]
  float* h_stats = agg + ND;  // [2][D]
  float* e_stats = h_stats + 2 * D;
  float* h_sc = e_stats + 2 * D;
  float* h_sh = h_sc + D;
  float* e_sc = h_sh + D;
  float* e_sh = e_sc + D;

  float* out = (float*)d_out;   // h_pre/h_out at [0, ND), e_pre/e_out at [ND, ND+ED)
  float* h_pre = out;
  float* e_pre = out + ND;

  // 1) zero agg + stats
  {
    const long zn = ND + 4 * D;
    const int blocks = (int)((zn + 255) / 256);
    gnn_zero_kernel<<<blocks, 256, 0, stream>>>(agg, zn);
  }
  // 2) node GEMMs (Uh, Vh, Bh, Ch)
  gnn_node_gemm_kernel<<<(N + 15) / 16, 128, 0, stream>>>(
      h_in, Uw, Vw, Bw, Cw, Ub, Vb, Bb, Cb, Uh, N);
  // 3) edge GEMM + gather + gated scatter + e stats
  gnn_edge_kernel<<<(E + 31) / 32, 256, 0, stream>>>(
      e_in, eidx, Aw, Ab, Vh, Bh, Ch, agg, e_stats, e_pre, E);
  // 4) h pre-BN + h stats
  gnn_node_post_kernel<<<(N + 15) / 16, 256, 0, stream>>>(Uh, agg, h_pre, h_stats, N);
  // 5) finalize BN params
  gnn_stats_final_kernel<<<1, 256, 0, stream>>>(
      h_stats, e_stats, h_gamma, h_beta, e_gamma, e_beta,
      h_sc, h_sh, e_sc, e_sh, N, E);
  // 6) apply BN + relu + residual in place
  {
    const long tot4 = (ND + ED) / 4;
    const int blocks = (int)((tot4 + 255) / 256);
    gnn_apply_kernel<<<blocks, 256, 0, stream>>>(
        h_in, e_in, h_sc, h_sh, e_sc, e_sh, out, ND, ED);
  }
}